// TemporalConvolutionLayer_40415642255479
// MI455X (gfx1250) — compile-verified
//
#include <hip/hip_runtime.h>
#include <math.h>

typedef float v2f __attribute__((ext_vector_type(2)));
typedef float v8f __attribute__((ext_vector_type(8)));

#define B_ 8
#define P_ 256
#define N_ 256
#define I_ 256
#define K_ 16
#define CH_ 40

// workspace layout (float offsets)
#define OFF_FRE 0u
#define OFF_FIM 65536u
#define OFF_ARE 131072u
#define OFF_AIM 196608u
#define OFF_YRE 262144u
#define OFF_YIM (262144u + 8388608u)

#define TWO_PI 6.28318530717958647692f

// ---------------------------------------------------------------------------
// Precompute DFT-256 matrix F[i][p] = exp(-2*pi*I * i*p/256), split planar.
__global__ void fill_F(float* __restrict__ ws) {
    int t = blockIdx.x * blockDim.x + threadIdx.x;   // 65536
    int i = t >> 8, p = t & 255;
    int ph = (i * p) & 255;                          // exact phase reduction
    float ang = -TWO_PI * (float)ph * (1.0f / 256.0f);
    float s, c;
    __sincosf(ang, &s, &c);
    ws[OFF_FRE + t] = c;
    ws[OFF_FIM + t] = s;
}

// ---------------------------------------------------------------------------
// Precompute A[i] = IDFT16 * W_h[i]   (fuses the ifft along j with the einsum)
// A[i][j][l] = (1/16) * sum_m exp(+2*pi*I*j*m/16) * W_h[i][m][l]
__global__ void compute_A(const float* __restrict__ wh, float* __restrict__ ws) {
    int i = blockIdx.x;            // 256 frequencies
    int t = threadIdx.x;           // 256 = j*16 + l
    int j = t >> 4, l = t & 15;
    float ar = 0.f, ai = 0.f;
    for (int m = 0; m < 16; ++m) {
        int ph = (j * m) & 15;
        float ang = TWO_PI * (float)ph * (1.0f / 16.0f);
        float s, c;
        __sincosf(ang, &s, &c);
        int wi = ((i * 16 + m) * 16 + l) * 2;        // complex64 interleaved
        float wr = wh[wi], wim = wh[wi + 1];
        ar += c * wr - s * wim;
        ai += c * wim + s * wr;
    }
    ws[OFF_ARE + (unsigned)i * 256 + t] = ar * (1.0f / 16.0f);
    ws[OFF_AIM + (unsigned)i * 256 + t] = ai * (1.0f / 16.0f);
}

// ---------------------------------------------------------------------------
// Stage 1: Y[b][i][n*16+l] = sum_p F[i][p] * x[b][p][n][l]   (DFT as GEMM)
// One wave per 16(i) x 16(l, fixed n) output tile; K=256 in 4-wide steps.
__global__ void __launch_bounds__(256) dft_stage(const float* __restrict__ x,
                                                 float* __restrict__ ws) {
    const int lane  = threadIdx.x & 31;
    const int wave  = threadIdx.x >> 5;
    const int tile  = blockIdx.x * 8 + wave;   // 0..32767 = b(8) * it(16) * n(256)
    const int b     = tile >> 12;
    const int it    = (tile >> 8) & 15;
    const int n     = tile & 255;

    const int row   = lane & 15;          // A-frag row / B-frag col
    const int khalf = (lane >> 4) << 1;   // 0 or 2 (K sub-block)
    const int half  = lane >> 4;

    const float* Fre = ws + OFF_FRE;
    const float* Fim = ws + OFF_FIM;
    const int i = it * 16 + row;

    v8f cRe = {}; v8f cIm = {};
    for (int p0 = 0; p0 < 256; p0 += 4) {
        const int pb = p0 + khalf;
        // A-frags: rows of F (re and im), cols p = pb, pb+1  (pb even -> v2f load)
        v2f aRe = *(const v2f*)(Fre + i * 256 + pb);
        v2f aIm = *(const v2f*)(Fim + i * 256 + pb);
        // B-frag: x[b][pb..pb+1][n][l], l = lane%16 (contiguous across lanes)
        size_t xi = (((size_t)b * P_ + pb) * N_ + n) * CH_ + row;
        v2f bx;
        bx.x = x[xi];
        bx.y = x[xi + (size_t)N_ * CH_];   // p+1
        cRe = __builtin_amdgcn_wmma_f32_16x16x4_f32(false, aRe, false, bx,
                                                    (short)0, cRe, false, false);
        cIm = __builtin_amdgcn_wmma_f32_16x16x4_f32(false, aIm, false, bx,
                                                    (short)0, cIm, false, false);
    }
    float* Yre = ws + OFF_YRE;
    float* Yim = ws + OFF_YIM;
#pragma unroll
    for (int r = 0; r < 8; ++r) {
        const int irow = it * 16 + r + half * 8;
        size_t yi = ((size_t)b * I_ + irow) * 4096 + n * 16 + row;
        Yre[yi] = cRe[r];
        Yim[yi] = cIm[r];
    }
}

// ---------------------------------------------------------------------------
// Stage 2: per (b,i): OUT[n][j] = A[i] * Y[b][i][n][:]   (complex 256x16x16 GEMM)
// then out[b][n][i][j] = x[b][n][i][j] + ctanh(OUT[n][j]) as complex64.
__global__ void __launch_bounds__(256) stage2(const float* __restrict__ x,
                                              float* __restrict__ out,
                                              const float* __restrict__ ws) {
    const int lane  = threadIdx.x & 31;
    const int wave  = threadIdx.x >> 5;
    const int b     = blockIdx.x >> 8;
    const int i     = blockIdx.x & 255;

    const float* Are = ws + OFF_ARE + (unsigned)i * 256;
    const float* Aim = ws + OFF_AIM + (unsigned)i * 256;
    const float* Yre = ws + OFF_YRE + ((size_t)b * I_ + i) * 4096;
    const float* Yim = ws + OFF_YIM + ((size_t)b * I_ + i) * 4096;

    const int row   = lane & 15;          // A-frag row (n) / B-frag col (j)
    const int khalf = (lane >> 4) << 1;   // K sub-block offset
    const int half  = lane >> 4;

    for (int t = wave; t < 16; t += 8) {          // 2 n-tiles per wave
        const int n0 = t * 16;
        v8f cRe = {}; v8f cIm = {};
#pragma unroll
        for (int kk = 0; kk < 16; kk += 4) {
            const int lc = kk + khalf;            // even -> aligned v2f loads
            v2f aRe = *(const v2f*)(Yre + (n0 + row) * 16 + lc);
            v2f aIm = *(const v2f*)(Yim + (n0 + row) * 16 + lc);
            // B = A[i]^T : b[l][j] = A[j][l]
            v2f bAr = *(const v2f*)(Are + row * 16 + lc);
            v2f bAi = *(const v2f*)(Aim + row * 16 + lc);
            v2f bAiN; bAiN.x = -bAi.x; bAiN.y = -bAi.y;
            // complex GEMM: Re += Yre*Ar - Yim*Ai ; Im += Yre*Ai + Yim*Ar
            cRe = __builtin_amdgcn_wmma_f32_16x16x4_f32(false, aRe, false, bAr,
                                                        (short)0, cRe, false, false);
            cRe = __builtin_amdgcn_wmma_f32_16x16x4_f32(false, aIm, false, bAiN,
                                                        (short)0, cRe, false, false);
            cIm = __builtin_amdgcn_wmma_f32_16x16x4_f32(false, aRe, false, bAi,
                                                        (short)0, cIm, false, false);
            cIm = __builtin_amdgcn_wmma_f32_16x16x4_f32(false, aIm, false, bAr,
                                                        (short)0, cIm, false, false);
        }
        // epilogue: complex tanh + residual, write interleaved complex64
#pragma unroll
        for (int r = 0; r < 8; ++r) {
            const int nn = n0 + r + half * 8;
            const int j  = row;
            float zr = cRe[r], zi = cIm[r];
            float a2 = fminf(fmaxf(2.0f * zr, -40.0f), 40.0f);
            float e  = __expf(a2);
            float ei = __frcp_rn(e);
            float sh = 0.5f * (e - ei);
            float chh = 0.5f * (e + ei);
            float s2, c2;
            __sincosf(2.0f * zi, &s2, &c2);
            float den = chh + c2;
            float tre = sh / den;
            float tim = s2 / den;
            size_t xi = (((size_t)b * P_ + nn) * N_ + i) * CH_ + j;
            float xv = x[xi];
            out[xi * 2]     = xv + tre;
            out[xi * 2 + 1] = tim;
        }
    }
}

// ---------------------------------------------------------------------------
// Channels 16..39: out = x + tanh(x) (real), imag = 0.
__global__ void zchan(const float* __restrict__ x, float* __restrict__ out) {
    size_t idx = (size_t)blockIdx.x * blockDim.x + threadIdx.x;
    const size_t total = (size_t)B_ * P_ * N_ * 24;
    if (idx >= total) return;
    int c = (int)(idx % 24);
    size_t r = idx / 24;                 // flattened (b,p,n)
    size_t xi = r * CH_ + 16 + c;
    float xv = x[xi];
    out[xi * 2]     = xv + tanhf(xv);
    out[xi * 2 + 1] = 0.0f;
}

// ---------------------------------------------------------------------------
extern "C" void kernel_launch(void* const* d_in, const int* in_sizes, int n_in,
                              void* d_out, int out_size, void* d_ws, size_t ws_size,
                              hipStream_t stream) {
    const float* x  = (const float*)d_in[0];   // (8,256,256,40) f32
    const float* wh = (const float*)d_in[1];   // (256,16,16) complex64 interleaved
    // d_in[2] (W_z) is dead code in the reference.
    float* out = (float*)d_out;                // complex64 interleaved
    float* ws  = (float*)d_ws;

    fill_F<<<256, 256, 0, stream>>>(ws);
    compute_A<<<256, 256, 0, stream>>>(wh, ws);
    dft_stage<<<4096, 256, 0, stream>>>(x, ws);
    stage2<<<2048, 256, 0, stream>>>(x, out, ws);
    const size_t nz = (size_t)B_ * P_ * N_ * 24;
    zchan<<<(unsigned)((nz + 255) / 256), 256, 0, stream>>>(x, out);
}